// LeakyRNN_2370821948033
// MI455X (gfx1250) — compile-verified
//
#include <hip/hip_runtime.h>
#include <cmath>

// Problem constants (match reference)
#define T_ 128
#define B_ 32
#define I_ 128
#define H_ 320
#define O_ 32
#define G_ 8
// GROUP = 16 (I_/G_), ALPHA_X = 0.2, ALPHA_W = 0.1

typedef __attribute__((ext_vector_type(2))) float v2f;
typedef __attribute__((ext_vector_type(8))) float v8f;

// mask_a / mask_o column factor: sign_h * exist.
// e_h = int(0.8*320) = 256 -> sign +1 for h<256 else -1
// zc  = int((1.0-0.8)*320) = int(63.9999...) = 63 -> exist 1 for h<257 else 0
__device__ __forceinline__ float maskAO(int h) {
    return (h < 256) ? 1.f : ((h == 256) ? -1.f : 0.f);
}
// mask_h column sign (no exist), diagonal handled separately
__device__ __forceinline__ float sgn(int h) { return (h < 256) ? 1.f : -1.f; }

// ---------------------------------------------------------------------------
// Kernel 0: (re)initialize per-batch plastic weights in workspace, transposed
//   wxT[b][i][h] = relu(W_x2h[h][i])                       (mask_x == 1)
//   whT[b][j][h] = relu(W_h2h[h][j]) * sgn(j) * (j != h)   (mask_h)
// ---------------------------------------------------------------------------
__global__ void init_weights(const float* __restrict__ Wx,
                             const float* __restrict__ Wh,
                             float* __restrict__ wxT,
                             float* __restrict__ whT) {
    const long total1 = (long)B_ * I_ * H_;
    const long total2 = (long)B_ * H_ * H_;
    const long stride = (long)gridDim.x * blockDim.x;
    for (long idx = (long)blockIdx.x * blockDim.x + threadIdx.x;
         idx < total1 + total2; idx += stride) {
        if (idx < total1) {
            int rem = (int)(idx % (I_ * H_));
            int i = rem / H_, h = rem % H_;
            wxT[idx] = fmaxf(Wx[h * I_ + i], 0.f);
        } else {
            long j2 = idx - total1;
            int rem = (int)(j2 % (H_ * H_));
            int j = rem / H_, h = rem % H_;
            float s = (j == h) ? 0.f : sgn(j);
            whT[j2] = fmaxf(Wh[h * H_ + j], 0.f) * s;
        }
    }
}

// ---------------------------------------------------------------------------
// Kernel 1: full recurrence, FUSED weight pass.
// One block per batch element (18.3 MB of per-batch weights stays hot in the
// 192 MB L2), one thread per hidden unit h (320 threads = 10 wave32s).
//
// Fusion: the plastic update that produces w_t (built from step t-1 values)
// is applied in the same streaming pass that accumulates the step-t mat-vec
// on the updated weights. This is legal because xm_t / pre_t depend only on
// output_t = no_{t-1}, available before the pass. The final (t = T-1) update
// is dropped entirely (reference discards final weights).
// Traffic: 1 read + 1 write per element per step vs 2 reads + 1 write.
// ---------------------------------------------------------------------------
__global__ __launch_bounds__(H_) void rnn_steps(
    const float* __restrict__ x,       // (T,B,I)
    const float* __restrict__ Rs,      // (T,B,1)
    const float* __restrict__ b_h2h,   // (H)
    const float* __restrict__ W_attn,  // (G,H)
    const float* __restrict__ b_attn,  // (G)
    const float* __restrict__ c_plas,  // (6)
    float* __restrict__ wxT,           // ws: (B,I,H)
    float* __restrict__ whT,           // ws: (B,H,H)
    float* __restrict__ hs)            // out: (T,B,H)
{
    const int b = blockIdx.x;
    const int h = threadIdx.x;  // 0..319
    const int wave = h >> 5, lane = h & 31;

    // Double-buffered activation vectors, indexed by t&1.
    __shared__ float sOutV[2][H_];  // sOutV[t&1] = output_t (presyn for matvec
                                    // at step t; presyn for update at step t+1)
    __shared__ float sPre[2][H_];   // sgn(j) * output_t[j]
    __shared__ float sXm[2][I_];    // attention-modulated input xm_t
    __shared__ float sLogit[G_];
    __shared__ float sAttn[G_];

    const float c0 = fabsf(c_plas[0]), c1 = fabsf(c_plas[1]), c2 = fabsf(c_plas[2]);
    const float c3 = fabsf(c_plas[3]), c4 = fabsf(c_plas[4]), c5 = fabsf(c_plas[5]);

    float state  = 0.f;
    float noPrev = 0.f;             // output_{t} of this thread's unit (no at t-1)
    sOutV[0][h] = 0.f;              // output_0 = relu(0)
    float* __restrict__ wxCol = wxT + (long)b * I_ * H_ + h;
    float* __restrict__ whCol = whT + (long)b * H_ * H_ + h;
    const float bias = b_h2h[h];
    __syncthreads();

    for (int t = 0; t < T_; ++t) {
        const int cur = t & 1, prv = cur ^ 1;

        // ---- attention logits from output_t: wave g computes logit g ------
        if (wave < G_) {
            float p = 0.f;
            #pragma unroll
            for (int k = 0; k < H_; k += 32) {
                int j = lane + k;
                p += sOutV[cur][j] * fmaxf(W_attn[wave * H_ + j], 0.f) * maskAO(j);
            }
            for (int off = 16; off; off >>= 1) p += __shfl_xor(p, off, 32);
            if (lane == 0) sLogit[wave] = p + b_attn[wave];
        }
        __syncthreads();

        // ---- softmax over 8 logits ----------------------------------------
        if (h == 0) {
            float m = sLogit[0];
            for (int g = 1; g < G_; ++g) m = fmaxf(m, sLogit[g]);
            float e[G_], s = 0.f;
            for (int g = 0; g < G_; ++g) { e[g] = __expf(sLogit[g] - m); s += e[g]; }
            float inv = 1.f / s;
            for (int g = 0; g < G_; ++g) sAttn[g] = e[g] * inv;
        }
        __syncthreads();

        // ---- xm_t and signed presyn vector for step t ---------------------
        if (h < I_)
            sXm[cur][h] = x[((long)t * B_ + b) * I_ + h] * sAttn[h >> 4] * (float)G_;
        sPre[cur][h] = sOutV[cur][h] * sgn(h);
        __syncthreads();

        // ---- fused pass: apply update t-1 -> w_t, accumulate mat-vec t ----
        float acc = bias;
        if (t == 0) {
            // First step: weights are w_0, read-only mat-vec.
            #pragma unroll 8
            for (int i = 0; i < I_; ++i) {
                if ((i & 31) == 0) __builtin_prefetch(&wxCol[(i + 32) * H_], 0, 0);
                acc += fmaxf(wxCol[i * H_], 0.f) * sXm[cur][i];
            }
            #pragma unroll 8
            for (int j = 0; j < H_; ++j) {
                if ((j & 31) == 0) __builtin_prefetch(&whCol[(j + 32) * H_], 0, 0);
                acc += fmaxf(whCol[j * H_], 0.f) * sPre[cur][j];
            }
            acc -= fmaxf(whCol[h * H_], 0.f) * sPre[cur][h];  // mask_h diagonal
        } else {
            // Update coefficients from step t-1 (reward R_{t-1}, post = noPrev)
            const float R  = Rs[(t - 1) * B_ + b];
            const float aR = 0.1f * R;                        // ALPHA_W * R
            const float px = aR * (c0 + c2 * noPrev), qx = aR * c1 * noPrev;
            const float ph = aR * (c3 + c5 * noPrev), qh = aR * c4 * noPrev;
            #pragma unroll 8
            for (int i = 0; i < I_; ++i) {
                if ((i & 31) == 0) __builtin_prefetch(&wxCol[(i + 32) * H_], 0, 0);
                float wn = wxCol[i * H_] * 0.9f + px * sXm[prv][i] + qx;
                wxCol[i * H_] = wn;
                acc += fmaxf(wn, 0.f) * sXm[cur][i];
            }
            #pragma unroll 8
            for (int j = 0; j < H_; ++j) {
                if ((j & 31) == 0) __builtin_prefetch(&whCol[(j + 32) * H_], 0, 0);
                float wn = whCol[j * H_] * 0.9f + ph * sOutV[prv][j] + qh;
                whCol[j * H_] = wn;
                acc += fmaxf(wn, 0.f) * sPre[cur][j];
            }
            // Remove the diagonal term (same-wave store->load is ordered)
            acc -= fmaxf(whCol[h * H_], 0.f) * sPre[cur][h];
        }

        // ---- leaky integration, nonlinearity ------------------------------
        state = state * 0.8f + acc * 0.2f;                    // ALPHA_X = 0.2
        const float no = tanhf(fmaxf(state, 0.f));
        hs[((long)t * B_ + b) * H_ + h] = no;

        // output_{t+1} = no overwrites the out_{t-1} slot; all reads of that
        // slot happened in the fused pass above -> barrier before the write.
        __syncthreads();
        sOutV[prv][h] = no;
        noPrev = no;
        __syncthreads();
    }
}

// ---------------------------------------------------------------------------
// Kernel 2: output projection as a real GEMM via V_WMMA_F32_16X16X4_F32.
//   out[m, o] = sigmoid( sum_k hs[m,k] * ew_o[o,k] + b_h2o[o] )
//   M = T*B = 4096, K = H = 320 (80 steps of 4), N = O = 32 (2 tiles)
// One wave per 16x16 tile. Fragment striping per ISA 7.12.2:
//   A (16x4):  VGPR r = elem(m = lane&15, k = r + (lane>=16 ? 2 : 0))
//   B (4x16):  VGPR r = elem(k = r + (lane>=16 ? 2 : 0), n = lane&15)
//   C (16x16): VGPR r = elem(m = r + (lane>=16 ? 8 : 0), n = lane&15)
// ---------------------------------------------------------------------------
__global__ __launch_bounds__(64) void project_wmma(
    const float* __restrict__ hs,      // (T*B, H)
    const float* __restrict__ W_h2o,   // (O, H)
    const float* __restrict__ b_h2o,   // (O)
    float* __restrict__ out)           // (T*B, O)
{
    const int lane  = threadIdx.x & 31;
    const int ntile = threadIdx.x >> 5;          // 0..1 (two 16-col tiles)
    const int mbase = blockIdx.x * 16;           // 256 blocks cover M=4096
    const int half  = (lane >= 16) ? 1 : 0;
    const int mrow  = mbase + (lane & 15);
    const int ncol  = ntile * 16 + (lane & 15);

    const float* __restrict__ arow = hs + (long)mrow * H_;
    const float* __restrict__ brow = W_h2o + (long)ncol * H_;

    v8f c = {};
    for (int k0 = 0; k0 < H_; k0 += 4) {
        const int ka = k0 + 2 * half;
        v2f a, bm;
        a.x  = arow[ka];
        a.y  = arow[ka + 1];
        bm.x = fmaxf(brow[ka], 0.f) * maskAO(ka);        // ew_o = relu(W)*mask_o
        bm.y = fmaxf(brow[ka + 1], 0.f) * maskAO(ka + 1);
        c = __builtin_amdgcn_wmma_f32_16x16x4_f32(
                /*neg_a=*/false, a, /*neg_b=*/false, bm,
                /*c_mod=*/(short)0, c, /*reuse_a=*/false, /*reuse_b=*/false);
    }

    const float bo = b_h2o[ncol];
    #pragma unroll
    for (int r = 0; r < 8; ++r) {
        const int row = mbase + r + half * 8;
        const float z = c[r] + bo;
        out[(long)row * O_ + ncol] = 1.f / (1.f + __expf(-z));
    }
}

// ---------------------------------------------------------------------------
extern "C" void kernel_launch(void* const* d_in, const int* in_sizes, int n_in,
                              void* d_out, int out_size, void* d_ws, size_t ws_size,
                              hipStream_t stream) {
    (void)in_sizes; (void)n_in; (void)out_size; (void)ws_size;
    const float* x      = (const float*)d_in[0];
    const float* Rs     = (const float*)d_in[1];
    const float* W_x2h  = (const float*)d_in[2];
    const float* W_h2h  = (const float*)d_in[3];
    const float* b_h2h  = (const float*)d_in[4];
    const float* W_h2o  = (const float*)d_in[5];
    const float* b_h2o  = (const float*)d_in[6];
    const float* W_attn = (const float*)d_in[7];
    const float* b_attn = (const float*)d_in[8];
    const float* c_plas = (const float*)d_in[9];

    float* out = (float*)d_out;                    // (T,B,O) first output
    float* hs  = out + (size_t)T_ * B_ * O_;       // (T,B,H) second output

    float* wxT = (float*)d_ws;                     // (B,I,H)
    float* whT = wxT + (size_t)B_ * I_ * H_;       // (B,H,H)

    init_weights<<<512, 256, 0, stream>>>(W_x2h, W_h2h, wxT, whT);
    rnn_steps<<<B_, H_, 0, stream>>>(x, Rs, b_h2h, W_attn, b_attn, c_plas,
                                     wxT, whT, hs);
    project_wmma<<<(T_ * B_) / 16, 64, 0, stream>>>(hs, W_h2o, b_h2o, out);
}